// OperatePrediction_25237227831919
// MI455X (gfx1250) — compile-verified
//
#include <hip/hip_runtime.h>

typedef __attribute__((ext_vector_type(2))) float v2f;
typedef __attribute__((ext_vector_type(4))) float v4f;
typedef __attribute__((ext_vector_type(8))) float v8f;

#define B_   16
#define S_   2048
#define H_   1024
#define NOP_ 4
#define RTOT (B_ * S_)                 // 32768 rows total
#define SCORES_ELEMS (B_ * S_ * NOP_)  // 131072 floats

// ---------------------------------------------------------------------------
// Kernel 1: per-batch stable compaction scan.
// One block per batch (16 blocks x 1024 threads, 2 elements/thread).
// Produces srcidx[b][j] = s of the j-th row with op_ids==0, and nvalid[b].
// ---------------------------------------------------------------------------
__global__ void __launch_bounds__(1024)
scan_kernel(const int* __restrict__ op_ids,
            int* __restrict__ srcidx,
            int* __restrict__ nvalid,
            const int* __restrict__ mu_p) {
    __shared__ int sc[1024];
    const int b = blockIdx.x;
    const int t = threadIdx.x;
    const int s0 = 2 * t, s1 = 2 * t + 1;
    const int f0 = (op_ids[b * S_ + s0] == 0) ? 1 : 0;
    const int f1 = (op_ids[b * S_ + s1] == 0) ? 1 : 0;
    sc[t] = f0 + f1;
    __syncthreads();
    // Hillis-Steele inclusive scan over 1024 per-thread counts
    for (int off = 1; off < 1024; off <<= 1) {
        int add = (t >= off) ? sc[t - off] : 0;
        __syncthreads();
        sc[t] += add;
        __syncthreads();
    }
    const int total = sc[1023];
    const int base  = sc[t] - f0 - f1;   // exclusive prefix for this thread
    const int mu    = *mu_p;
    if (f0 && (base < mu))        srcidx[b * S_ + base]      = s0;
    if (f1 && (base + f0 < mu))   srcidx[b * S_ + base + f0] = s1;
    if (t == 0) nvalid[b] = (total < mu) ? total : mu;
}

// ---------------------------------------------------------------------------
// Kernel 2: state_scores = state @ W^T + b via V_WMMA_F32_16X16X4_F32.
// 4 waves/block, each wave owns one 16-row tile; K-loop of 256 WMMAs.
// W^T staged in LDS as [k][n] padded to n=16 (64 KB) -> conflict-free reads.
// ---------------------------------------------------------------------------
__global__ void __launch_bounds__(128)
scores_kernel(const float* __restrict__ state,
              const float* __restrict__ W,
              const float* __restrict__ bias,
              float* __restrict__ out) {
    __shared__ float WtT[H_ * 16];     // WtT[k*16 + n] = (n<4) ? W[n][k] : 0
    for (int i = threadIdx.x; i < H_ * 16; i += 128) {
        const int n = i & 15;
        const int k = i >> 4;
        WtT[i] = (n < NOP_) ? W[n * H_ + k] : 0.0f;
    }
    __syncthreads();

    const int wave = threadIdx.x >> 5;
    const int lane = threadIdx.x & 31;
    const int tile = blockIdx.x * 4 + wave;
    const int row0 = tile * 16;
    const int m    = lane & 15;        // A: M index / B: N index / C: N index
    const int hi   = lane >> 4;        // lane-half selects K pair {0,1} vs {2,3}

    // A-fragment source: row (row0+m), starting at K offset 2*hi (8B aligned)
    const float* arow = state + (size_t)(row0 + m) * H_ + 2 * hi;

    v8f c = {0.f, 0.f, 0.f, 0.f, 0.f, 0.f, 0.f, 0.f};
    #pragma unroll 8
    for (int k = 0; k < H_; k += 4) {
        v2f a = *(const v2f*)(arow + k);          // A[m][kk], A[m][kk+1]
        const int kk = k + 2 * hi;
        v2f bb;
        bb.x = WtT[kk * 16 + m];                  // B[kk][n]   = W[n][kk]
        bb.y = WtT[(kk + 1) * 16 + m];            // B[kk+1][n] = W[n][kk+1]
        c = __builtin_amdgcn_wmma_f32_16x16x4_f32(
                /*neg_a=*/false, a, /*neg_b=*/false, bb,
                /*c_mod=*/(short)0, c, /*reuse_a=*/false, /*reuse_b=*/false);
    }

    // C/D layout: VGPR r holds M = r + 8*hi, N = lane&15. Only N<4 is real.
    const float bv = (m < NOP_) ? bias[m] : 0.0f;
    if (m < NOP_) {
        #pragma unroll
        for (int r = 0; r < 8; ++r) {
            out[(size_t)(row0 + r + 8 * hi) * NOP_ + m] = c[r] + bv;
        }
    }
}

// ---------------------------------------------------------------------------
// Kernel 3: decoder_inputs gather + zero pad. One block per (b, j) row,
// 256 threads x 16B = full 1024-float row, nontemporal streaming store.
// ---------------------------------------------------------------------------
__global__ void __launch_bounds__(256)
gather_kernel(const float* __restrict__ state,
              const int* __restrict__ srcidx,
              const int* __restrict__ nvalid,
              const int* __restrict__ mu_p,
              float* __restrict__ dec) {
    const int b  = blockIdx.x >> 11;        // / 2048
    const int j  = blockIdx.x & (S_ - 1);
    const int mu = *mu_p;
    if (j >= mu) return;
    const int nv = nvalid[b];
    const int col = threadIdx.x * 4;

    v4f v = {0.f, 0.f, 0.f, 0.f};
    if (j < nv) {
        const int s = srcidx[b * S_ + j];
        v = *(const v4f*)(state + ((size_t)b * S_ + s) * H_ + col);
    }
    v4f* dst = (v4f*)(dec + ((size_t)b * mu + j) * H_ + col);
    __builtin_nontemporal_store(v, dst);
}

// ---------------------------------------------------------------------------
extern "C" void kernel_launch(void* const* d_in, const int* in_sizes, int n_in,
                              void* d_out, int out_size, void* d_ws, size_t ws_size,
                              hipStream_t stream) {
    const float* state  = (const float*)d_in[0];
    // d_in[1] = input_ids (unused by the reference outputs)
    const int*   op_ids = (const int*)d_in[2];
    const float* W      = (const float*)d_in[3];
    const float* bias   = (const float*)d_in[4];
    const int*   mu_p   = (const int*)d_in[5];
    float* out = (float*)d_out;

    int* srcidx = (int*)d_ws;            // B_*S_ ints
    int* nvalid = srcidx + B_ * S_;      // B_ ints

    // 1) stable compaction indices per batch
    scan_kernel<<<B_, 1024, 0, stream>>>(op_ids, srcidx, nvalid, mu_p);

    // 2) WMMA scores: 32768 rows / 16 per wave / 4 waves per block = 512 blocks
    scores_kernel<<<RTOT / 16 / 4, 128, 0, stream>>>(state, W, bias, out);

    // 3) gather + zero-pad decoder inputs
    gather_kernel<<<B_ * S_, 256, 0, stream>>>(state, srcidx, nvalid, mu_p,
                                               out + SCORES_ELEMS);
}